// ScalarDotProductCriticNetwork_49658411876916
// MI455X (gfx1250) — compile-verified
//
#include <hip/hip_runtime.h>
#include <hip/hip_bf16.h>

// MI455X (gfx1250) fused critic network.
// Algebraic reduction: av[b,i,j] has only 2 distinct rows per j, so
//   wav[b,a,i] = base[b,a] + weight[b,a,i]*diff[b,i]
//   h[b,a,i]   = leaky(U[b,a] + weight[b,a,i]*V[b,i]),  value = h @ W_f2
// Total ~3.3 GFLOP fp32, 35.7 MB output -> output-BW bound; fp32 WMMA
// (v_wmma_f32_16x16x4_f32) used for every GEMM to keep reference precision.

typedef float v2f __attribute__((ext_vector_type(2)));
typedef float v8f __attribute__((ext_vector_type(8)));

#define NWAVES 8
#define BS 132   // padded row stride for [32][128] LDS tiles (132%64==4 -> conflict-free)
#define SS 33    // padded stride for [32][32] tiles
#define US 68    // padded stride for [32][64] tiles

// Accumulate a 16x16 fp32 tile: A element (m=lane%16, k), B element (k, n=lane%16).
// A: Aptr[m*aRs + k].  B: Bptr[k*bKs + n*bNs]  (bKs/bNs let us read B or B^T).
// Fragment layout per CDNA5 ISA 7.12.2: lane L holds K = 4*kstep + 2*(L/16) + v.
__device__ __forceinline__ v8f wmma_acc_k(const float* A, int aRs,
                                          const float* B, int bKs, int bNs,
                                          int K, v8f acc) {
  const int lane = threadIdx.x & 31;
  const int r    = lane & 15;
  const int koff = (lane >> 4) << 1;
  const float* Ap = A + r * aRs + koff;
  const float* Bp = B + r * bNs + koff * bKs;
  for (int k0 = 0; k0 < K; k0 += 4) {
    v2f a, b;
    a.x = Ap[k0];
    a.y = Ap[k0 + 1];
    b.x = Bp[k0 * bKs];
    b.y = Bp[(k0 + 1) * bKs];
    acc = __builtin_amdgcn_wmma_f32_16x16x4_f32(false, a, false, b,
                                                (short)0, acc, false, false);
  }
  return acc;
}

// Store C/D fragment: lane L, VGPR v -> (M = v + 8*(L/16), N = L%16).
__device__ __forceinline__ void store_tile(float* D, int dRs, int mt16, int nt16,
                                           v8f acc, float scale, int do_tanh) {
  const int lane = threadIdx.x & 31;
  const int col  = nt16 + (lane & 15);
  const int half = lane >> 4;
#pragma unroll
  for (int v = 0; v < 8; ++v) {
    int row = mt16 + v + 8 * half;
    float x = acc[v] * scale;
    if (do_tanh) x = tanhf(x);
    D[row * dRs + col] = x;
  }
}

__global__ __launch_bounds__(256)
void critic_fused_kernel(const float* __restrict__ states,
                         const float* __restrict__ policies,
                         const float* __restrict__ actions,
                         const float* __restrict__ states_p,
                         const float* __restrict__ actions_p,
                         const float* __restrict__ Wk,  const float* __restrict__ Wq,
                         const float* __restrict__ Wv,  const float* __restrict__ Wkp,
                         const float* __restrict__ Wqp, const float* __restrict__ Wvp,
                         const float* __restrict__ Wf1, const float* __restrict__ Wf2,
                         float* __restrict__ out) {
  extern __shared__ float sm[];
  float* S0  = sm;              // oa                       [32][BS]
  float* S1  = S0 + 32 * BS;    // op -> kp -> wavp
  float* S2  = S1 + 32 * BS;    // oa_p
  float* S3  = S2 + 32 * BS;    // q -> av_oa -> qp -> avp
  float* S4  = S3 + 32 * BS;    // k -> av_op -> diff
  float* S5  = S4 + 32 * BS;    // base
  float* UL  = S5 + 32 * BS;    // U [32][US]
  float* VL  = UL + 32 * US;    // V [32][US]
  float* W2L = VL + 32 * US;    // W_f2 [64][SS]
  float* wL  = W2L + 64 * SS;   // weight [32][SS]
  float* pL  = wL + 32 * SS;    // wp [32][SS]
  float* sL  = pL + 32 * SS;    // scores scratch [32][SS]

  const int b    = blockIdx.x;
  const int tid  = threadIdx.x;
  const int wave = tid >> 5;
  const int lane = tid & 31;
  const float inv_sqrt_dk = 0.08838834764831845f;  // 1/sqrt(128)

  const float* st  = states    + (size_t)b * 32 * 96;
  const float* po  = policies  + (size_t)b * 32 * 32;
  const float* ac  = actions   + (size_t)b * 32 * 32;
  const float* stp = states_p  + (size_t)b * 32 * 96;
  const float* acp = actions_p + (size_t)b * 32 * 32;

  // ---- Load: oa=[states|actions], op=[states|policies], oa_p=[sp|ap], W_f2 ----
  for (int idx = tid; idx < 4096; idx += 256) {
    int n = idx >> 7, c = idx & 127;
    if (c < 96) {
      float sv = st[n * 96 + c];
      S0[n * BS + c] = sv;
      S1[n * BS + c] = sv;
      S2[n * BS + c] = stp[n * 96 + c];
    } else {
      int c2 = c - 96;
      S0[n * BS + c] = ac[n * 32 + c2];
      S1[n * BS + c] = po[n * 32 + c2];
      S2[n * BS + c] = acp[n * 32 + c2];
    }
  }
  for (int idx = tid; idx < 2048; idx += 256)
    W2L[(idx >> 5) * SS + (idx & 31)] = Wf2[idx];
  __syncthreads();

  // ---- G1: q = oa@Wq -> S3 ; k = oa@Wk -> S4  (32 tiles, K=128) ----
  for (int t = wave; t < 32; t += NWAVES) {
    int g = t >> 4, tt = t & 15, mt = tt >> 3, nt = tt & 7;
    v8f acc = {0, 0, 0, 0, 0, 0, 0, 0};
    acc = wmma_acc_k(S0 + mt * 16 * BS, BS, (g ? Wk : Wq) + nt * 16, 128, 1, 128, acc);
    store_tile(g ? S4 : S3, BS, mt * 16, nt * 16, acc, 1.f, 0);
  }
  __syncthreads();

  // ---- scores^T: s[a][j] = k[a].q[j]  (B = q^T via strides) ----
  for (int t = wave; t < 4; t += NWAVES) {
    int mt = t >> 1, nt = t & 1;
    v8f acc = {0, 0, 0, 0, 0, 0, 0, 0};
    acc = wmma_acc_k(S4 + mt * 16 * BS, BS, S3 + nt * 16 * BS, 1, BS, 128, acc);
    store_tile(sL, SS, mt * 16, nt * 16, acc, 1.f, 0);
  }
  __syncthreads();

  // ---- row softmax -> weight; write weight output ----
  if (tid < 32) {
    int a = tid;
    float mx = -1e30f;
    for (int j = 0; j < 32; ++j) mx = fmaxf(mx, sL[a * SS + j]);
    float sum = 0.f;
    for (int j = 0; j < 32; ++j) {
      float e = expf((sL[a * SS + j] - mx) * inv_sqrt_dk);
      wL[a * SS + j] = e;
      sum += e;
    }
    float inv = 1.f / sum;
    float* ow = out + 8388608 + (size_t)b * 1024 + a * 32;
    for (int j = 0; j < 32; ++j) {
      float w = wL[a * SS + j] * inv;
      wL[a * SS + j] = w;
      ow[j] = w;
    }
  }
  __syncthreads();

  // ---- G2: av_oa = tanh(oa@Wv) -> S3 ; av_op = tanh(op@Wv) -> S4 ----
  for (int t = wave; t < 32; t += NWAVES) {
    int g = t >> 4, tt = t & 15, mt = tt >> 3, nt = tt & 7;
    v8f acc = {0, 0, 0, 0, 0, 0, 0, 0};
    acc = wmma_acc_k((g ? S1 : S0) + mt * 16 * BS, BS, Wv + nt * 16, 128, 1, 128, acc);
    store_tile(g ? S4 : S3, BS, mt * 16, nt * 16, acc, 1.f, 1);
  }
  __syncthreads();

  // ---- base = weight@av_oa / 32 -> S5  (K=32) ----
  for (int t = wave; t < 16; t += NWAVES) {
    int mt = t >> 3, nt = t & 7;
    v8f acc = {0, 0, 0, 0, 0, 0, 0, 0};
    acc = wmma_acc_k(wL + mt * 16 * SS, SS, S3 + nt * 16, BS, 1, 32, acc);
    store_tile(S5, BS, mt * 16, nt * 16, acc, 0.03125f, 0);
  }
  __syncthreads();

  // ---- diff = (av_op - av_oa)/32 -> S4 (in place) ----
  for (int idx = tid; idx < 4096; idx += 256) {
    int n = idx >> 7, c = idx & 127;
    S4[n * BS + c] = (S4[n * BS + c] - S3[n * BS + c]) * 0.03125f;
  }
  __syncthreads();

  // ---- G4: qp = oa@Wqp -> S3 ; kp = oa_p@Wkp -> S1 ----
  for (int t = wave; t < 32; t += NWAVES) {
    int g = t >> 4, tt = t & 15, mt = tt >> 3, nt = tt & 7;
    v8f acc = {0, 0, 0, 0, 0, 0, 0, 0};
    acc = wmma_acc_k((g ? S2 : S0) + mt * 16 * BS, BS,
                     (g ? Wkp : Wqp) + nt * 16, 128, 1, 128, acc);
    store_tile(g ? S1 : S3, BS, mt * 16, nt * 16, acc, 1.f, 0);
  }
  __syncthreads();

  // ---- scores_p[n][p] = qp[n].kp[p] ----
  for (int t = wave; t < 4; t += NWAVES) {
    int mt = t >> 1, nt = t & 1;
    v8f acc = {0, 0, 0, 0, 0, 0, 0, 0};
    acc = wmma_acc_k(S3 + mt * 16 * BS, BS, S1 + nt * 16 * BS, 1, BS, 128, acc);
    store_tile(sL, SS, mt * 16, nt * 16, acc, 1.f, 0);
  }
  __syncthreads();

  // ---- column softmax (over n) -> wp; write wp output ----
  if (tid < 32) {
    int p = tid;
    float mx = -1e30f;
    for (int n = 0; n < 32; ++n) mx = fmaxf(mx, sL[n * SS + p]);
    float sum = 0.f;
    for (int n = 0; n < 32; ++n) {
      float e = expf((sL[n * SS + p] - mx) * inv_sqrt_dk);
      pL[n * SS + p] = e;
      sum += e;
    }
    float inv = 1.f / sum;
    float* ow = out + 8650752 + (size_t)b * 1024;
    for (int n = 0; n < 32; ++n) {
      float w = pL[n * SS + p] * inv;
      pL[n * SS + p] = w;
      ow[n * 32 + p] = w;
    }
  }
  __syncthreads();

  // ---- avp = tanh(oa_p@Wvp) -> S3 ----
  for (int t = wave; t < 16; t += NWAVES) {
    int mt = t >> 3, nt = t & 7;
    v8f acc = {0, 0, 0, 0, 0, 0, 0, 0};
    acc = wmma_acc_k(S2 + mt * 16 * BS, BS, Wvp + nt * 16, 128, 1, 128, acc);
    store_tile(S3, BS, mt * 16, nt * 16, acc, 1.f, 1);
  }
  __syncthreads();

  // ---- wavp = wp@avp / 32 -> S1  (K=32) ----
  for (int t = wave; t < 16; t += NWAVES) {
    int mt = t >> 3, nt = t & 7;
    v8f acc = {0, 0, 0, 0, 0, 0, 0, 0};
    acc = wmma_acc_k(pL + mt * 16 * SS, SS, S3 + nt * 16, BS, 1, 32, acc);
    store_tile(S1, BS, mt * 16, nt * 16, acc, 0.03125f, 0);
  }
  __syncthreads();

  // ---- U = base@Wf1[:128] + wavp@Wf1[128:] -> UL ; V = diff@Wf1[:128] -> VL ----
  for (int t = wave; t < 16; t += NWAVES) {
    int isV = t >> 3, tt = t & 7, mt = tt >> 2, nt = tt & 3;
    v8f acc = {0, 0, 0, 0, 0, 0, 0, 0};
    if (isV) {
      acc = wmma_acc_k(S4 + mt * 16 * BS, BS, Wf1 + nt * 16, 64, 1, 128, acc);
      store_tile(VL, US, mt * 16, nt * 16, acc, 1.f, 0);
    } else {
      acc = wmma_acc_k(S5 + mt * 16 * BS, BS, Wf1 + nt * 16, 64, 1, 128, acc);
      acc = wmma_acc_k(S1 + mt * 16 * BS, BS, Wf1 + 128 * 64 + nt * 16, 64, 1, 128, acc);
      store_tile(UL, US, mt * 16, nt * 16, acc, 1.f, 0);
    }
  }
  __syncthreads();

  // ---- Stage 2: value[b,a,i,:] = leaky(U[a] + weight[a,i]*V[i]) @ W_f2 ----
  // 1024 (a,i) rows x 32 cols per batch = 64 row-tiles x 2 col-tiles.
  float* vout = out + (size_t)b * 32768;
  for (int t = wave; t < 128; t += NWAVES) {
    int ct = t & 1;
    int rt = t >> 1;
    int a  = rt >> 1;             // constant within tile
    int ib = (rt & 1) << 4;       // i = ib + m
    int r    = lane & 15;
    int koff = (lane >> 4) << 1;
    int iA   = ib + r;
    float w  = wL[a * SS + iA];
    const float* Urow = UL + a * US;
    const float* Vrow = VL + iA * US;
    v8f acc = {0, 0, 0, 0, 0, 0, 0, 0};
#pragma unroll
    for (int k0 = 0; k0 < 64; k0 += 4) {
      int kk = k0 + koff;
      float hx = Urow[kk] + w * Vrow[kk];
      float hy = Urow[kk + 1] + w * Vrow[kk + 1];
      hx = hx > 0.f ? hx : 0.01f * hx;
      hy = hy > 0.f ? hy : 0.01f * hy;
      v2f av, bv;
      av.x = hx;
      av.y = hy;
      bv.x = W2L[kk * SS + ct * 16 + r];
      bv.y = W2L[(kk + 1) * SS + ct * 16 + r];
      acc = __builtin_amdgcn_wmma_f32_16x16x4_f32(false, av, false, bv,
                                                  (short)0, acc, false, false);
    }
    int half = lane >> 4;
    int col  = ct * 16 + r;
#pragma unroll
    for (int v = 0; v < 8; ++v) {
      int iD = ib + v + 8 * half;
      vout[(size_t)((a * 32 + iD) * 32) + col] = acc[v];
    }
  }
}

extern "C" void kernel_launch(void* const* d_in, const int* in_sizes, int n_in,
                              void* d_out, int out_size, void* d_ws, size_t ws_size,
                              hipStream_t stream) {
  (void)in_sizes; (void)n_in; (void)out_size; (void)d_ws; (void)ws_size;
  const float* states    = (const float*)d_in[0];
  const float* policies  = (const float*)d_in[1];
  const float* actions   = (const float*)d_in[2];
  const float* states_p  = (const float*)d_in[3];
  const float* actions_p = (const float*)d_in[4];
  const float* Wk  = (const float*)d_in[5];
  const float* Wq  = (const float*)d_in[6];
  const float* Wv  = (const float*)d_in[7];
  const float* Wkp = (const float*)d_in[8];
  const float* Wqp = (const float*)d_in[9];
  const float* Wvp = (const float*)d_in[10];
  const float* Wf1 = (const float*)d_in[11];
  const float* Wf2 = (const float*)d_in[12];
  float* out = (float*)d_out;

  const size_t shmem =
      (size_t)(6 * 32 * BS + 2 * 32 * US + 64 * SS + 3 * 32 * SS) * sizeof(float);
  critic_fused_kernel<<<256, 256, shmem, stream>>>(
      states, policies, actions, states_p, actions_p,
      Wk, Wq, Wv, Wkp, Wqp, Wvp, Wf1, Wf2, out);
}